// ClusteredGraphReconstructor_31318901522601
// MI455X (gfx1250) — compile-verified
//
#include <hip/hip_runtime.h>
#include <math.h>

// MI455X / gfx1250 implementation.
//
// Strategy: hoist the bilinear form. P[r] = A @ W[r] via
// V_WMMA_F32_16X16X4_F32 (fp32 matrix pipe, exact), then each edge is a
// 64-length dot of two L2-resident rows (P+A = 64MB << 192MB L2), wave32
// one-edge-per-wave with fully coalesced float2 lane loads.
// Loss reduction is fully deterministic (block partials + fixed-order tree).

typedef __attribute__((ext_vector_type(2))) float v2f;
typedef __attribute__((ext_vector_type(8))) float v8f;

#define K_DIM 64
#define EDGE_BLOCKS 4096   // fixed grid for edge kernels -> fixed partial count

__device__ __forceinline__ float sigmoid_f(float x) {
    return 1.0f / (1.0f + __expf(-x));
}

// ---------------------------------------------------------------------------
// Kernel 1: W[r] = sigmoid(logits) * clamp(sigmoid(log_alpha)*1.2 - 0.1, 0, 1)
// ---------------------------------------------------------------------------
__global__ void build_weights_kernel(const float* __restrict__ logits,
                                     const float* __restrict__ log_alpha,
                                     float* __restrict__ Wout,
                                     int total) {
    int i = blockIdx.x * blockDim.x + threadIdx.x;
    if (i < total) {
        float g = sigmoid_f(log_alpha[i]) * 1.2f + (-0.1f);
        g = fminf(fmaxf(g, 0.0f), 1.0f);
        Wout[i] = sigmoid_f(logits[i]) * g;
    }
}

// ---------------------------------------------------------------------------
// Kernel 2: P[r] = A (N x 64) @ W[r] (64 x 64), fp32 WMMA 16x16x4.
// One wave per 16x16 output tile.  All 16 A-pairs (global_load_b64,
// contiguous) and 16 B-pairs are preloaded into registers, then 16
// accumulating WMMAs issue back-to-back behind a single load wait.
// Row index is clamped (not element-guarded) so loads are branch-free.
//
// Fragment layouts per CDNA5 ISA:
//  A (16x4 f32): lanes 0-15 hold M=lane, {v0,v1}={K=k0,k0+1};
//                lanes 16-31 hold M=lane-16, {v0,v1}={K=k0+2,k0+3}.
//  B (4x16 f32): v0: lanes0-15 = row k0 (N=0..15), lanes16-31 = row k0+2;
//                v1: lanes0-15 = row k0+1,         lanes16-31 = row k0+3.
//  C/D (16x16):  VGPR v, lanes 0-15 -> M=v, lanes 16-31 -> M=v+8; N=lane&15.
// ---------------------------------------------------------------------------
__global__ void gemm_P_kernel(const float* __restrict__ A,
                              const float* __restrict__ W,
                              float* __restrict__ P,
                              int N) {
    const int lane  = threadIdx.x;      // 0..31 (wave32)
    const int col   = lane & 15;        // M index for A-frag, N index for B/C
    const int khalf = lane >> 4;        // which K pair / which M half for C
    const int m0 = blockIdx.x * 16;
    const int n0 = blockIdx.y * 16;
    const int r  = blockIdx.z;

    const float* Wr = W + (size_t)r * K_DIM * K_DIM;
    const int arow  = min(m0 + col, N - 1);           // clamp: branch-free loads
    const float* ap = A + (size_t)arow * K_DIM;
    const float* bp = Wr + 2 * khalf * K_DIM + n0 + col;

    v2f afrag[16];
    v2f bfrag[16];
#pragma unroll
    for (int t = 0; t < 16; ++t) {
        // A pair is contiguous (8B aligned) -> one global_load_b64.
        afrag[t] = *(const v2f*)(ap + 4 * t + 2 * khalf);
        v2f b;
        b.x = bp[(size_t)(4 * t + 0) * K_DIM];
        b.y = bp[(size_t)(4 * t + 1) * K_DIM];
        bfrag[t] = b;
    }

    v8f acc = {};
#pragma unroll
    for (int t = 0; t < 16; ++t) {
        // 8 args: (neg_a, A, neg_b, B, c_mod, C, reuse_a, reuse_b)
        acc = __builtin_amdgcn_wmma_f32_16x16x4_f32(
            false, afrag[t], false, bfrag[t], (short)0, acc, false, false);
    }

    float* Pr = P + (size_t)r * N * K_DIM;
#pragma unroll
    for (int v = 0; v < 8; ++v) {
        const int row = m0 + v + 8 * khalf;
        if (row < N) Pr[(size_t)row * K_DIM + n0 + col] = acc[v];
    }
}

// ---------------------------------------------------------------------------
// Kernel 3: per-edge logits + softplus block partials.
// One edge per wave32: lane l loads float2 of P[r][src] and A[dst]
// (coalesced 256B row reads, L2-resident), dot, shfl_xor reduction.
// sign = -1 for positive edges (softplus(-logit)), +1 for negative edges.
// Each block writes one partial sum (deterministic, no float atomics).
// ---------------------------------------------------------------------------
__global__ void edge_logits_kernel(const float* __restrict__ P,
                                   const float* __restrict__ Anodes,
                                   const float* __restrict__ bias,
                                   const long long* __restrict__ eidx,   // [2,E]
                                   const long long* __restrict__ etype,  // [E]
                                   float* __restrict__ out_logits,
                                   float* __restrict__ partials,         // [grid]
                                   long long E, int N, float sign) {
    const int lane = threadIdx.x & 31;
    const int wid  = threadIdx.x >> 5;
    const int wpb  = blockDim.x >> 5;
    long long gw = (long long)blockIdx.x * wpb + wid;
    long long nw = (long long)gridDim.x * wpb;

    float local = 0.0f;
    for (long long e = gw; e < E; e += nw) {
        const long long s = eidx[e];
        const long long d = eidx[E + e];
        const long long r = etype[e];
        const float2* prow =
            (const float2*)(P + ((size_t)r * (size_t)N + (size_t)s) * K_DIM);
        const float2* drow = (const float2*)(Anodes + (size_t)d * K_DIM);
        const float2 pv = prow[lane];
        const float2 dv = drow[lane];
        float sum = fmaf(pv.x, dv.x, pv.y * dv.y);
#pragma unroll
        for (int off = 16; off > 0; off >>= 1)
            sum += __shfl_xor(sum, off, 32);
        if (lane == 0) {
            const float logit = sum + bias[r];
            out_logits[e] = logit;
            const float x = sign * logit;
            // numerically stable softplus
            local += fmaxf(x, 0.0f) + log1pf(__expf(-fabsf(x)));
        }
    }

    __shared__ float red[32];
    if (lane == 0) red[wid] = local;
    __syncthreads();
    if (threadIdx.x == 0) {
        float s = 0.0f;
        for (int i = 0; i < wpb; ++i) s += red[i];
        partials[blockIdx.x] = s;   // unconditional: every block writes
    }
}

// ---------------------------------------------------------------------------
// Kernel 4: recon_loss = (sum(pos partials) + sum(neg partials)) / E
// Single block, fixed-order strided + tree reduction -> deterministic.
// ---------------------------------------------------------------------------
__global__ void finalize_kernel(const float* __restrict__ ppos,
                                const float* __restrict__ pneg,
                                float* __restrict__ out,
                                int nblocks, float invE) {
    __shared__ float s[256];
    float a = 0.0f;
    for (int i = threadIdx.x; i < nblocks; i += 256)
        a += ppos[i] + pneg[i];
    s[threadIdx.x] = a;
    __syncthreads();
    for (int st = 128; st > 0; st >>= 1) {
        if (threadIdx.x < st) s[threadIdx.x] += s[threadIdx.x + st];
        __syncthreads();
    }
    if (threadIdx.x == 0) out[0] = s[0] * invE;
}

// ---------------------------------------------------------------------------
extern "C" void kernel_launch(void* const* d_in, const int* in_sizes, int n_in,
                              void* d_out, int out_size, void* d_ws, size_t ws_size,
                              hipStream_t stream) {
    (void)n_in; (void)out_size; (void)ws_size;

    const float*      assignments = (const float*)d_in[0];       // [N,64]
    const float*      icl         = (const float*)d_in[1];       // [R,64,64]
    const float*      lalpha      = (const float*)d_in[2];       // [R,64,64]
    const float*      bias        = (const float*)d_in[3];       // [R]
    const long long*  eidx        = (const long long*)d_in[4];   // [2,E] i64
    const long long*  etype       = (const long long*)d_in[5];   // [E]   i64
    const long long*  neidx       = (const long long*)d_in[6];   // [2,E] i64
    const long long*  netype      = (const long long*)d_in[7];   // [E]   i64
    float* out = (float*)d_out;  // [0]=recon, [1..E]=pos, [1+E..2E]=neg

    const int       Wtotal = in_sizes[1];            // R*64*64
    const int       R      = in_sizes[3];
    const long long E      = (long long)in_sizes[5];
    const int       N      = in_sizes[0] / K_DIM;

    // Workspace layout (256B aligned sections):
    //   [ partials_pos (EDGE_BLOCKS f32) | partials_neg (EDGE_BLOCKS f32)
    //     | W (Wtotal f32) | P (R*N*64 f32) ]
    char*  ws   = (char*)d_ws;
    float* ppos = (float*)ws;
    float* pneg = ppos + EDGE_BLOCKS;
    float* Wmat = pneg + EDGE_BLOCKS;
    float* P    = Wmat + Wtotal;

    // 1) gates + weights
    build_weights_kernel<<<(Wtotal + 255) / 256, 256, 0, stream>>>(
        icl, lalpha, Wmat, Wtotal);

    // 2) P[r] = A @ W[r] via fp32 WMMA
    dim3 gridG((N + 15) / 16, K_DIM / 16, R);
    gemm_P_kernel<<<gridG, 32, 0, stream>>>(assignments, Wmat, P, N);

    // 3) edge scoring, one wave per edge, grid-stride, fixed grid
    edge_logits_kernel<<<EDGE_BLOCKS, 256, 0, stream>>>(
        P, assignments, bias, eidx, etype, out + 1, ppos, E, N, -1.0f);
    edge_logits_kernel<<<EDGE_BLOCKS, 256, 0, stream>>>(
        P, assignments, bias, neidx, netype, out + 1 + E, pneg, E, N, 1.0f);

    // 4) scalar loss (deterministic reduction)
    finalize_kernel<<<1, 256, 0, stream>>>(ppos, pneg, out, EDGE_BLOCKS,
                                           1.0f / (float)E);
}